// TttM1BMMTritonModule_35210141892673
// MI455X (gfx1250) — compile-verified
//
#include <hip/hip_runtime.h>
#include <stdint.h>

// TTT-M1 fast-weight decode step, MI455X (gfx1250, wave32).
// One workgroup (256 threads = 8 wave32) per (b,h). Memory-bound:
// 64KB of W traffic per (b,h), ~256MB total -> ~11us at 23.3 TB/s.

typedef __attribute__((ext_vector_type(2))) float v2f;
typedef __attribute__((ext_vector_type(8))) float v8f;
typedef __attribute__((ext_vector_type(4))) int   v4i;

#define HF   64
#define BH   4096              // B*H = 128*32
#define MAT  4096              // HF*HF
#define OFF_WI (BH * HF)                 // Out elements: 262144
#define OFF_WG (OFF_WI + BH * MAT)       // + W1_init elements

#define AS1 __attribute__((address_space(1)))
#define AS3 __attribute__((address_space(3)))

#if defined(__gfx1250__) && __has_builtin(__builtin_amdgcn_global_load_async_to_lds_b128)
#define USE_ASYNC_LDS 1
#else
#define USE_ASYNC_LDS 0
#endif

#if USE_ASYNC_LDS
static __device__ __forceinline__ AS3 v4i* lds_cast(void* p) {
  // generic LDS pointer low 32 bits == LDS byte offset on gfx1250
  return (AS3 v4i*)(uint32_t)(uintptr_t)p;
}
static __device__ __forceinline__ AS1 v4i* glob_cast(const void* p) {
  return (AS1 v4i*)(uintptr_t)p;   // builtin wants non-const AS1 int4*
}
static __device__ __forceinline__ void async_wait0() {
#if __has_builtin(__builtin_amdgcn_s_wait_asynccnt)
  __builtin_amdgcn_s_wait_asynccnt(0);
#else
  asm volatile("s_wait_asynccnt 0" ::: "memory");
#endif
}
#endif

__global__ __launch_bounds__(256) void ttt_m1_decode(
    const float* __restrict__ Wi, const float* __restrict__ Wg,
    const float* __restrict__ XA, const float* __restrict__ XB,
    const float* __restrict__ XC, const float* __restrict__ Coef,
    float* __restrict__ OutO, float* __restrict__ WiO, float* __restrict__ WgO)
{
  __shared__ __attribute__((aligned(16))) float sWi[MAT];   // 16 KB
  __shared__ __attribute__((aligned(16))) float sWg[MAT];   // 16 KB
  __shared__ float sXB[HF], sXC[HF];
  __shared__ float sZp[2 * HF];       // two k-half partials of XB@W1_init
  __shared__ float sOutP[4 * HF];     // Out partials (4 row-groups)

  const int  t     = threadIdx.x;
  const int  bh    = blockIdx.x;
  const long base  = (long)bh * MAT;
  const long vbase = (long)bh * HF;

  // ---- Phase 0: stage W1_init / W1_grad into LDS ----
#if USE_ASYNC_LDS
  {
#pragma unroll
    for (int j = 0; j < 4; ++j) {
      const int idx = (t + 256 * j) * 4;  // 16B chunk per thread per iter
      __builtin_amdgcn_global_load_async_to_lds_b128(
          glob_cast(Wi + base + idx), lds_cast(&sWi[idx]), 0, 0);
      __builtin_amdgcn_global_load_async_to_lds_b128(
          glob_cast(Wg + base + idx), lds_cast(&sWg[idx]), 0, 0);
    }
  }
#else
  {
#pragma unroll
    for (int j = 0; j < 16; ++j) {
      const int idx = t + 256 * j;
      sWi[idx] = Wi[base + idx];
      sWg[idx] = Wg[base + idx];
    }
  }
#endif
  if (t < HF)            sXB[t]      = XB[vbase + t];
  else if (t < 2 * HF)   sXC[t - HF] = XC[vbase + t - HF];
#if USE_ASYNC_LDS
  async_wait0();
#endif
  __syncthreads();

  // ---- Phase 1: Z = XB @ W1_init via chained V_WMMA_F32_16X16X4_F32 ----
  // A[m][k] = XB[kbase+k0+k] broadcast over m (documented A layout: k = vgpr + 2*half).
  // B[k][n] = W1_init[kbase+k0+k][n0+n] (4x16 tile from LDS).
  // D row 0 (c[0], lanes 0..15) = partial (XB @ W1_init)[n0 + lane].
  {
    const int wave  = t >> 5;
    const int lane  = t & 31;
    const int h2    = (lane >> 4) << 1;     // 2*half
    const int ln    = lane & 15;
    const int n0    = (wave & 3) << 4;      // which 16 output columns
    const int kbase = (wave >> 2) << 5;     // which 32-row k-half
    v8f c = {0.f, 0.f, 0.f, 0.f, 0.f, 0.f, 0.f, 0.f};
#pragma unroll
    for (int k0 = 0; k0 < 32; k0 += 4) {
      v2f a, b;
      a.x = sXB[kbase + k0 + h2];
      a.y = sXB[kbase + k0 + 1 + h2];
      b.x = sWi[(kbase + k0 + h2)     * HF + n0 + ln];
      b.y = sWi[(kbase + k0 + 1 + h2) * HF + n0 + ln];
      c = __builtin_amdgcn_wmma_f32_16x16x4_f32(
          false, a, false, b, (short)0, c, false, false);
    }
    if (lane < 16) sZp[(wave >> 2) * HF + n0 + lane] = c[0];
  }
  __syncthreads();

  // ---- Phase 2: elementwise update + Out partials (single streaming pass) ----
  const int   d     = t & 63;                         // column, constant per thread
  const float coeff = Coef[bh];
  const float z1    = sZp[d] + sZp[HF + d] - XA[vbase + d];
  float po = 0.f;
#pragma unroll
  for (int i = 0; i < 16; ++i) {
    const int   e = t + 256 * i;                      // coalesced
    const int   f = e >> 6;                           // row
    const float g = sWg[e] + sXB[f] * z1;             // W1_grad_new
    const float w = sWi[e] - coeff * g;               // W1_init_new
    WgO[base + e] = g;
    WiO[base + e] = w;
    po += sXC[f] * w;                                 // Out partial
  }
  sOutP[(t >> 6) * HF + d] = po;
  __syncthreads();
  if (t < HF) {
    OutO[vbase + t] =
        sOutP[t] + sOutP[HF + t] + sOutP[2 * HF + t] + sOutP[3 * HF + t];
  }
}

extern "C" void kernel_launch(void* const* d_in, const int* in_sizes, int n_in,
                              void* d_out, int out_size, void* d_ws, size_t ws_size,
                              hipStream_t stream) {
  const float* Wi = (const float*)d_in[0];
  const float* Wg = (const float*)d_in[1];
  const float* XA = (const float*)d_in[2];
  const float* XB = (const float*)d_in[3];
  const float* XC = (const float*)d_in[4];
  const float* Cf = (const float*)d_in[5];
  float* out = (float*)d_out;
  ttt_m1_decode<<<BH, 256, 0, stream>>>(Wi, Wg, XA, XB, XC, Cf,
                                        out, out + OFF_WI, out + OFF_WG);
}